// MappingModule_17738214932564
// MI455X (gfx1250) — compile-verified
//
#include <hip/hip_runtime.h>

// ---------------------------------------------------------------------------
// Point-cloud -> BEV semantic grid scatter-max for MI455X (gfx1250, wave32).
// Memory/atomic bound: ~92 MB traffic -> ~4us floor @ 23.3 TB/s. No GEMM
// structure, so no WMMA; CDNA5 paths used: TDM tensor_load_to_lds (+
// s_wait_tensorcnt) for LDS param staging, NT temporal hints on the
// single-use point stream, non-returning global_atomic_max_u32 into L2.
// ---------------------------------------------------------------------------

typedef unsigned int       u32;
typedef unsigned long long u64;
typedef __attribute__((ext_vector_type(4))) float v4f;
typedef __attribute__((ext_vector_type(4))) int   v4i;
typedef __attribute__((ext_vector_type(4))) u32   v4u;
typedef __attribute__((ext_vector_type(8))) int   v8i;

namespace {
constexpr int   kB       = 8;
constexpr int   kRows    = 480;          // ceil(24.0 / 0.05)
constexpr int   kCols    = 480;
constexpr int   kCells   = kRows * kCols;      // 230400
constexpr float kDhMin   = 1.25f;
constexpr float kDhMax   = 0.75f;
constexpr float kHalfMap = 12.0f;        // H_M/2 == W_M/2
constexpr float kRes     = 0.05f;
}

#define USE_TDM 1

// --- kernel 1: zero the dense output grid (b128 stores) ---------------------
__global__ void zero_out_kernel(v4f* __restrict__ out, int n4) {
    int i = blockIdx.x * blockDim.x + threadIdx.x;
    if (i < n4) out[i] = (v4f){0.f, 0.f, 0.f, 0.f};
}

// --- kernel 2: per-batch params: cos, sin, px, pz, hmin, hmax (8 floats) ----
__global__ void precompute_params_kernel(const float* __restrict__ pose_start,
                                         const float* __restrict__ heading_start,
                                         const float* __restrict__ current_height,
                                         float* __restrict__ params) {
    int b = threadIdx.x;
    if (b < kB) {
        float ang = -heading_start[b];
        params[8 * b + 0] = cosf(ang);
        params[8 * b + 1] = sinf(ang);
        params[8 * b + 2] = pose_start[3 * b + 0];
        params[8 * b + 3] = pose_start[3 * b + 2];
        params[8 * b + 4] = current_height[b] - kDhMin;
        params[8 * b + 5] = current_height[b] + kDhMax;
        params[8 * b + 6] = 0.f;
        params[8 * b + 7] = 0.f;
    }
}

// --- kernel 3: stream points (4/thread), scatter-max into grid --------------
__launch_bounds__(256)
__global__ void scatter_max_kernel(const float* __restrict__ xyz,
                                   const int*   __restrict__ batch_idx,
                                   const int*   __restrict__ semantics,
                                   const float* __restrict__ params,
                                   u32*         __restrict__ out,
                                   int n_points, int n_threads) {
    __shared__ float sp[8 * kB];   // 256 B: 8 batches x 8 floats

#if defined(USE_TDM) && __has_builtin(__builtin_amdgcn_tensor_load_to_lds)
    // TDM DMA of the 256-byte parameter table into LDS (wave 0 issues it).
    // D# per CDNA5 ISA sec 8.3/8.4: count=1, type=2(image), data_size=4B,
    // tensor 64x1, tile 64x1, stride 64 elements. 6-arg builtin form:
    // (v4u group0, v8i group1, v4i group2, v4i group3, v8i extra, i32 cpol)
    if ((threadIdx.x >> 5) == 0) {
        u64 ga  = (u64)(uintptr_t)params;
        u32 lds = (u32)(uintptr_t)sp;      // low 32 bits of LDS aperture addr
        v4u g0;
        g0[0] = 1u;                                   // count=1, user mode
        g0[1] = lds;                                  // lds_addr
        g0[2] = (u32)(ga & 0xFFFFFFFFu);              // global_addr[31:0]
        g0[3] = (u32)((ga >> 32) & 0x1FFFFFFu) | (2u << 30); // addr[56:32]|type=2
        v8i g1;
        g1[0] = (int)(2u << 16);    // workgroup_mask=0, data_size=2 (4 bytes)
        g1[1] = (int)(64u << 16);   // tensor_dim0 = 64 elements
        g1[2] = (int)(1u  << 16);   // tensor_dim1 = 1
        g1[3] = (int)(64u << 16);   // tile_dim0 = 64
        g1[4] = 1;                  // tile_dim1 = 1
        g1[5] = 64;                 // tensor_dim0_stride = 64
        g1[6] = (int)(64u << 16);   // tensor_dim1_stride = 64
        g1[7] = 0;
        v4i z4 = {0, 0, 0, 0};
        v8i z8 = {0, 0, 0, 0, 0, 0, 0, 0};
        __builtin_amdgcn_tensor_load_to_lds(g0, g1, z4, z4, z8, 0);
        __builtin_amdgcn_s_wait_tensorcnt(0);
    }
#else
    if (threadIdx.x < 8 * kB) sp[threadIdx.x] = params[threadIdx.x];
#endif
    __syncthreads();

    int t = blockIdx.x * blockDim.x + threadIdx.x;
    if (t >= n_threads) return;
    int base = t << 2;                     // 4 points per thread

    float px[4], py[4], pz[4];
    int   bis[4], sms[4];

    if (base + 3 < n_points) {
        // 48B of xyz = exactly three b128 loads; NT hint (single-use stream).
        const v4f* x4 = (const v4f*)(xyz + (size_t)base * 3);
        v4f a = __builtin_nontemporal_load(x4 + 0);
        v4f b = __builtin_nontemporal_load(x4 + 1);
        v4f c = __builtin_nontemporal_load(x4 + 2);
        v4i bi = __builtin_nontemporal_load((const v4i*)(batch_idx + base));
        v4i sm = __builtin_nontemporal_load((const v4i*)(semantics + base));
        px[0] = a.x; px[1] = a.w; px[2] = b.z; px[3] = c.y;
        py[0] = a.y; py[1] = b.x; py[2] = b.w; py[3] = c.z;
        pz[0] = a.z; pz[1] = b.y; pz[2] = c.x; pz[3] = c.w;
        bis[0] = bi.x; bis[1] = bi.y; bis[2] = bi.z; bis[3] = bi.w;
        sms[0] = sm.x; sms[1] = sm.y; sms[2] = sm.z; sms[3] = sm.w;
    } else {
#pragma unroll
        for (int j = 0; j < 4; ++j) {
            int i = base + j;
            bool live = i < n_points;
            int k = live ? i : 0;
            px[j] = xyz[3 * k + 0];
            py[j] = xyz[3 * k + 1];
            pz[j] = xyz[3 * k + 2];
            bis[j] = batch_idx[k];
            sms[j] = live ? semantics[k] : 0;   // sem=0 -> skipped below
        }
    }

#pragma unroll
    for (int j = 0; j < 4; ++j) {
        int bb = bis[j] & (kB - 1);
        const float* pb = &sp[bb << 3];
        float cc = pb[0], ss = pb[1];
        float qx = px[j] - pb[2];
        float qz = pz[j] - pb[3];
        float rx = cc * qx + ss * qz;          // x' =  c*px + s*pz
        float rz = cc * qz - ss * qx;          // z' = -s*px + c*pz
        int row = __float2int_rn((rz + kHalfMap) / kRes);
        int col = __float2int_rn((rx + kHalfMap) / kRes);
        bool ok = (py[j] > pb[4]) & (py[j] < pb[5]) &
                  (row >= 0) & (row < kRows) & (col >= 0) & (col < kCols) &
                  (sms[j] > 0);                // max with 0.0 is a no-op
        if (ok) {
            // non-negative float bit patterns order as u32 -> fire-and-forget
            // global_atomic_max_u32 resolved in L2 (7.4 MB target << 192 MB).
            u32 v = __float_as_uint((float)sms[j]);
            atomicMax(out + (size_t)bb * kCells + row * kCols + col, v);
        }
    }
}

extern "C" void kernel_launch(void* const* d_in, const int* in_sizes, int n_in,
                              void* d_out, int out_size, void* d_ws, size_t ws_size,
                              hipStream_t stream) {
    const float* xyz            = (const float*)d_in[0];
    const float* pose_start     = (const float*)d_in[1];
    const float* heading_start  = (const float*)d_in[2];
    const float* current_height = (const float*)d_in[3];
    const int*   batch_indices  = (const int*)d_in[4];
    const int*   semantics      = (const int*)d_in[5];
    float*       params         = (float*)d_ws;           // 256 B scratch

    int n_points = in_sizes[0] / 3;                       // 4,194,304
    int n4       = out_size / 4;                          // out_size divisible by 4

    hipLaunchKernelGGL(zero_out_kernel, dim3((n4 + 255) / 256), dim3(256), 0,
                       stream, (v4f*)d_out, n4);
    hipLaunchKernelGGL(precompute_params_kernel, dim3(1), dim3(kB), 0, stream,
                       pose_start, heading_start, current_height, params);

    int n_threads = (n_points + 3) / 4;                   // 1,048,576
    hipLaunchKernelGGL(scatter_max_kernel, dim3((n_threads + 255) / 256),
                       dim3(256), 0, stream, xyz, batch_indices, semantics,
                       params, (u32*)d_out, n_points, n_threads);
}